// PoseEstimator_75857712382174
// MI455X (gfx1250) — compile-verified
//
#include <hip/hip_runtime.h>
#include <math.h>

// ---------------------------------------------------------------------------
// Pose estimator objective for MI455X (gfx1250, wave32).
// Heavy part: symmetric Chamfer over P=4 pairs of 8192-point clouds.
// Strategy:
//   d2(i,j) = |x_i|^2 + A_i . B_j   with A_i = (x0,x1,x2,1),
//                                        B_j = (-2y0,-2y1,-2y2,|y|^2)
//   -> one V_WMMA_F32_16X16X4_F32 per 16x16 tile, epilogue = 8 v_min per tile.
//   min_j sqrt(d2) == sqrt(min_j d2): sqrt deferred to the reduction.
//   Column-direction handled by a second pass with A/B roles swapped.
// ---------------------------------------------------------------------------

#define NPARTS 4
#define NPTS   8192
#define NTILES (NPTS / 16)          // 512 tiles of 16 points per part

typedef float v2f __attribute__((ext_vector_type(2)));
typedef float v8f __attribute__((ext_vector_type(8)));

// ---- workspace layout (bytes) ----------------------------------------------
#define WS_T      0                                   // transforms: 4*16 f32
#define WS_ACC    512                                 // 16 doubles (geo[4][2], dist)
#define WS_X2CAD  4096                                // |x|^2 per cad pt  [P][N] f32
#define WS_X2CAM  (WS_X2CAD + NPARTS * NPTS * 4)      // |y|^2 per cam pt
#define WS_ACAD   (WS_X2CAM + NPARTS * NPTS * 4)      // A-layout, transformed cad
#define WS_BCAD   (WS_ACAD + NPARTS * NPTS * 16)      // B-layout, transformed cad
#define WS_ACAM   (WS_BCAD + NPARTS * NPTS * 16)      // A-layout, camera
#define WS_BCAM   (WS_ACAM + NPARTS * NPTS * 16)      // B-layout, camera
// total ~2.3 MB

// ---------------------------------------------------------------------------
// Kernel 1: zero accumulators + build per-part 4x4 transforms from quats.
// ---------------------------------------------------------------------------
__global__ void pe_init(const float* __restrict__ rot_quats,
                        const float* __restrict__ tras,
                        char* __restrict__ ws) {
  int t = threadIdx.x;
  double* acc = (double*)(ws + WS_ACC);
  if (t < 16) acc[t] = 0.0;
  if (t < NPARTS) {
    const float* q = rot_quats + 4 * t;
    float w = q[0], x = q[1], y = q[2], z = q[3];
    float inv = 1.0f / sqrtf(w * w + x * x + y * y + z * z);
    w *= inv; x *= inv; y *= inv; z *= inv;
    float* Tp = (float*)(ws + WS_T) + 16 * t;
    Tp[0]  = 1.f - 2.f*y*y - 2.f*z*z; Tp[1]  = 2.f*x*y - 2.f*w*z; Tp[2]  = 2.f*w*y + 2.f*x*z; Tp[3]  = tras[3*t + 0];
    Tp[4]  = 2.f*x*y + 2.f*w*z; Tp[5]  = 1.f - 2.f*x*x - 2.f*z*z; Tp[6]  = 2.f*y*z - 2.f*w*x; Tp[7]  = tras[3*t + 1];
    Tp[8]  = 2.f*x*z - 2.f*w*y; Tp[9]  = 2.f*w*x + 2.f*y*z; Tp[10] = 1.f - 2.f*x*x - 2.f*y*y; Tp[11] = tras[3*t + 2];
    Tp[12] = 0.f; Tp[13] = 0.f; Tp[14] = 0.f; Tp[15] = 1.f;
  }
}

// ---------------------------------------------------------------------------
// Kernel 2: transform cad points, pre-swizzle A/B operand streams into the
// exact WMMA f32 16x16x4 lane layouts (tile-major, 256 B per 16-row tile):
//   A tile: lane l(0..15) -> (K0,K1) of row l; lane l+16 -> (K2,K3) of row l.
//   B tile: mirrored K-split per lane half, column = l & 15.
// Also accumulates the pointwise-distance term for E_kin.
// ---------------------------------------------------------------------------
__global__ __launch_bounds__(256) void pe_prep(const float* __restrict__ cam,
                                               const float* __restrict__ cad,
                                               char* __restrict__ ws) {
  const int p = blockIdx.y;
  const int n = blockIdx.x * 256 + threadIdx.x;
  const float* T = (const float*)(ws + WS_T) + 16 * p;

  const float* cd = cad + ((size_t)p * NPTS + n) * 3;
  float c0 = cd[0], c1 = cd[1], c2 = cd[2];
  float x0 = T[0] * c0 + T[1] * c1 + T[2]  * c2 + T[3];
  float x1 = T[4] * c0 + T[5] * c1 + T[6]  * c2 + T[7];
  float x2 = T[8] * c0 + T[9] * c1 + T[10] * c2 + T[11];

  const float* cm = cam + ((size_t)p * NPTS + n) * 3;
  float y0 = cm[0], y1 = cm[1], y2 = cm[2];

  float xs = x0 * x0 + x1 * x1 + x2 * x2;
  float ys = y0 * y0 + y1 * y1 + y2 * y2;
  ((float*)(ws + WS_X2CAD))[p * NPTS + n] = xs;
  ((float*)(ws + WS_X2CAM))[p * NPTS + n] = ys;

  const int t = n >> 4, lm = n & 15;
  const size_t base = (size_t)p * NPTS * 4 + (size_t)t * 64;  // in floats
  float* Acad = (float*)(ws + WS_ACAD) + base;
  float* Bcad = (float*)(ws + WS_BCAD) + base;
  float* Acam = (float*)(ws + WS_ACAM) + base;
  float* Bcam = (float*)(ws + WS_BCAM) + base;

  Acad[lm*2 + 0] = x0;        Acad[lm*2 + 1] = x1;
  Acad[(lm+16)*2 + 0] = x2;   Acad[(lm+16)*2 + 1] = 1.0f;
  Bcad[lm*2 + 0] = -2.f*x0;   Bcad[lm*2 + 1] = -2.f*x1;
  Bcad[(lm+16)*2 + 0] = -2.f*x2; Bcad[(lm+16)*2 + 1] = xs;
  Acam[lm*2 + 0] = y0;        Acam[lm*2 + 1] = y1;
  Acam[(lm+16)*2 + 0] = y2;   Acam[(lm+16)*2 + 1] = 1.0f;
  Bcam[lm*2 + 0] = -2.f*y0;   Bcam[lm*2 + 1] = -2.f*y1;
  Bcam[(lm+16)*2 + 0] = -2.f*y2; Bcam[(lm+16)*2 + 1] = ys;

  // E_kin "distances" term: mean_{p,n} ||x_t - scam|| (homog comps cancel)
  float dx = x0 - y0, dy = x1 - y1, dz = x2 - y2;
  double dn = sqrt((double)(dx * dx + dy * dy + dz * dz));
  #pragma unroll
  for (int off = 16; off; off >>= 1) dn += __shfl_xor(dn, off, 32);
  if ((threadIdx.x & 31) == 0)
    atomicAdd((double*)(ws + WS_ACC) + 8, dn);
}

// ---------------------------------------------------------------------------
// Kernel 3: Chamfer one-direction row-min pass.
//   dir 0: rows = transformed cad, cols = camera  (min over axis 1)
//   dir 1: rows = camera, cols = transformed cad  (min over axis 0)
// One wave per 16-row tile; B stream staged through LDS in 16 KB chunks.
// Inner loop: 1 ds b64 + 1 v_wmma_f32_16x16x4_f32 + 8 v_min per y-tile.
// ---------------------------------------------------------------------------
__global__ __launch_bounds__(256) void pe_chamfer(char* __restrict__ ws) {
  const int p    = blockIdx.y;
  const int dir  = blockIdx.z;
  const int lane = threadIdx.x & 31;
  const int wave = threadIdx.x >> 5;
  const int xt   = blockIdx.x * 8 + wave;

  const char* Abase = ws + (dir == 0 ? WS_ACAD : WS_ACAM);
  const char* Bbase = ws + (dir == 0 ? WS_BCAM : WS_BCAD);
  const float* x2a  = (const float*)(ws + (dir == 0 ? WS_X2CAD : WS_X2CAM)) + p * NPTS;
  double* acc = (double*)(ws + WS_ACC) + p * 2 + dir;

  // A tile for this wave: one coalesced b64 per lane, reused for all y.
  const v2f* Ap = (const v2f*)(Abase + (size_t)p * NPTS * 16 + (size_t)xt * 256);
  v2f a = Ap[lane];

  const v2f* Bg = (const v2f*)(Bbase + (size_t)p * NPTS * 16);
  __shared__ v2f sB[64 * 32];  // 64 y-tiles = 16 KB

  v8f rowmin;
  #pragma unroll
  for (int k = 0; k < 8; ++k) rowmin[k] = 3.0e38f;

  for (int chunk = 0; chunk < 8; ++chunk) {
    __syncthreads();
    #pragma unroll
    for (int i = 0; i < 8; ++i)
      sB[threadIdx.x + 256 * i] = Bg[chunk * 2048 + threadIdx.x + 256 * i];
    if (chunk + 1 < 8)  // pull next chunk toward the caches (global_prefetch)
      __builtin_prefetch((const void*)(Bg + (chunk + 1) * 2048 + threadIdx.x * 8), 0, 0);
    __syncthreads();

    #pragma unroll 4
    for (int yt = 0; yt < 64; ++yt) {
      v2f b = sB[yt * 32 + lane];
      v8f cz = {};
      // D = A x B (+0): 16x16 tile of (-2 x.y + |y|^2)
      v8f d = __builtin_amdgcn_wmma_f32_16x16x4_f32(
          /*neg_a=*/false, a, /*neg_b=*/false, b,
          /*c_mod=*/(short)0, cz, /*reuse_a=*/false, /*reuse_b=*/false);
      #pragma unroll
      for (int k = 0; k < 8; ++k) rowmin[k] = __builtin_fminf(rowmin[k], d[k]);
    }
  }

  // Min across the 16 columns held by one lane-half (wave32, 16-wide xor).
  #pragma unroll
  for (int k = 0; k < 8; ++k) {
    float m = rowmin[k];
    m = fminf(m, __shfl_xor(m, 1, 16));
    m = fminf(m, __shfl_xor(m, 2, 16));
    m = fminf(m, __shfl_xor(m, 4, 16));
    m = fminf(m, __shfl_xor(m, 8, 16));
    rowmin[k] = m;
  }

  // C/D layout: VGPR k, lanes 0-15 -> M=k; lanes 16-31 -> M=k+8.
  if (lane == 0 || lane == 16) {
    const int half = lane >> 4;
    double s = 0.0;
    #pragma unroll
    for (int k = 0; k < 8; ++k) {
      float d2 = rowmin[k] + x2a[xt * 16 + half * 8 + k];
      d2 = fmaxf(d2, 0.0f);
      s += sqrt((double)d2);
    }
    atomicAdd(acc, s);
  }
}

// ---------------------------------------------------------------------------
// Kernel 4: scalar f64 tail — eners, softmax, E_kin, outputs.
// out = [all_obj, errors(5), transforms(64)] as f32 (70 elems).
// ---------------------------------------------------------------------------
__device__ inline void mat4vec(const float* M, const float* v, float* r) {
  #pragma unroll
  for (int i = 0; i < 4; ++i)
    r[i] = M[i*4+0]*v[0] + M[i*4+1]*v[1] + M[i*4+2]*v[2] + M[i*4+3]*v[3];
}

__global__ void pe_final(const float* __restrict__ part_weight,
                         const float* __restrict__ rt_k,
                         const float* __restrict__ joint_axes,
                         const char* __restrict__ ws,
                         float* __restrict__ out) {
  const float* T = (const float*)(ws + WS_T);
  if (threadIdx.x < 64) out[6 + threadIdx.x] = T[threadIdx.x];
  if (threadIdx.x != 0) return;

  const double* acc = (const double*)(ws + WS_ACC);
  double eners[NPARTS];
  #pragma unroll
  for (int p = 0; p < NPARTS; ++p)
    eners[p] = 5.0 * (acc[p*2 + 0] / (double)NPTS + acc[p*2 + 1] / (double)NPTS);

  double mn = eners[0], mx = eners[0];
  for (int p = 1; p < NPARTS; ++p) { mn = fmin(mn, eners[p]); mx = fmax(mx, eners[p]); }
  double et[NPARTS], emx = -1e300;
  for (int p = 0; p < NPARTS; ++p) { et[p] = (eners[p] - mn) / (mx - mn + 1e-8); emx = fmax(emx, et[p]); }
  double sm[NPARTS], den = 0.0;
  for (int p = 0; p < NPARTS; ++p) { sm[p] = exp(et[p] - emx); den += sm[p]; }
  for (int p = 0; p < NPARTS; ++p) sm[p] /= den;

  double errors[NPARTS], all_obj = 0.0, emax = -1e300;
  for (int p = 0; p < NPARTS; ++p) {
    errors[p] = (double)part_weight[p] * sm[p] * eners[p];
    all_obj += sm[p] * eners[p];
    emax = fmax(emax, errors[p]);
  }

  double distances = acc[8] / (double)(NPARTS * NPTS);

  // Tj_q = rt_k[0] @ T0 @ joint_axes[0,0]
  float u[4], Tj[4];
  mat4vec(T, joint_axes + 0, u);
  mat4vec(rt_k, u, Tj);
  double ss = 0.0;
  for (int p = 1; p < NPARTS; ++p) {
    float v[4], w[4];
    mat4vec(T + 16 * p, joint_axes + ((p - 1) * 2 + 1) * 4, v);
    mat4vec(rt_k + 16 * p, v, w);
    for (int i = 0; i < 3; ++i) {
      double d = (double)Tj[i] - (double)w[i];
      ss += d * d;
    }
  }
  double nrm = sqrt(ss) / (double)NPARTS;
  double e_kin = log(0.1 * nrm + 1.0) + distances;
  double e_kin_total = emax * e_kin;
  all_obj += e_kin_total;

  out[0] = (float)all_obj;
  for (int p = 0; p < NPARTS; ++p) out[1 + p] = (float)errors[p];
  out[5] = (float)e_kin_total;
}

// ---------------------------------------------------------------------------
extern "C" void kernel_launch(void* const* d_in, const int* in_sizes, int n_in,
                              void* d_out, int out_size, void* d_ws, size_t ws_size,
                              hipStream_t stream) {
  const float* camera_pts  = (const float*)d_in[0];
  const float* cad_pts     = (const float*)d_in[1];
  const float* part_weight = (const float*)d_in[2];
  const float* rot_quats   = (const float*)d_in[3];
  const float* tras        = (const float*)d_in[4];
  const float* rt_k        = (const float*)d_in[5];
  const float* joint_axes  = (const float*)d_in[6];
  char* ws = (char*)d_ws;
  float* out = (float*)d_out;

  pe_init<<<1, 64, 0, stream>>>(rot_quats, tras, ws);
  pe_prep<<<dim3(NPTS / 256, NPARTS), 256, 0, stream>>>(camera_pts, cad_pts, ws);
  pe_chamfer<<<dim3(NTILES / 8, NPARTS, 2), 256, 0, stream>>>(ws);
  pe_final<<<1, 64, 0, stream>>>(part_weight, rt_k, joint_axes, ws, out);
}